// LUTBlock_52364241273392
// MI455X (gfx1250) — compile-verified
//
#include <hip/hip_runtime.h>
#include <stdint.h>

// Problem constants (match reference)
#define B_SZ   16384
#define F_SZ   512
#define T_SZ   16
#define C_SZ   8
#define ROWS_SZ 256
#define OUT_SZ 512

#define RB      8      // batch rows staged per block
#define THREADS 256    // 8 waves (wave32)

typedef float float4v __attribute__((ext_vector_type(4)));
typedef int   int4v   __attribute__((__vector_size__(4 * sizeof(int))));

#define AS1 __attribute__((address_space(1)))
#define AS3 __attribute__((address_space(3)))

#if __has_builtin(__builtin_amdgcn_global_load_async_to_lds_b128)
#define HAVE_ASYNC_LDS 1
#else
#define HAVE_ASYNC_LDS 0
#endif

__global__ __launch_bounds__(THREADS) void lut_block_kernel(
    const float* __restrict__ x,       // [B, F]
    const float* __restrict__ table,   // [T, ROWS, OUT]
    const int*   __restrict__ aa,      // [T, C]
    const int*   __restrict__ ab,      // [T, C]
    float*       __restrict__ y)       // [B, OUT]
{
    __shared__ float xs[RB][F_SZ];     // 16 KB staged x rows
    __shared__ int   sidx[RB][T_SZ];   // packed LUT indices

    const int tid = threadIdx.x;
    const int b0  = blockIdx.x * RB;

    // ---------------- Phase 0: stage 8 x-rows (16 KB) into LDS ----------------
    {
        const float* src = x + (size_t)b0 * F_SZ;   // 4096 contiguous floats
        float*       dst = &xs[0][0];
#if HAVE_ASYNC_LDS
        #pragma unroll
        for (int i = 0; i < 4; ++i) {
            const int e = (tid + i * THREADS) * 4;  // float4 granularity
            AS1 int4v* gp = (AS1 int4v*)(uintptr_t)(const void*)(src + e);
            AS3 int4v* lp = (AS3 int4v*)(uint32_t)(uintptr_t)(void*)(dst + e);
            __builtin_amdgcn_global_load_async_to_lds_b128(gp, lp, 0, 0);
        }
#if __has_builtin(__builtin_amdgcn_s_wait_asynccnt)
        __builtin_amdgcn_s_wait_asynccnt(0);
#else
        asm volatile("s_wait_asynccnt 0" ::: "memory");
#endif
#else
        #pragma unroll
        for (int i = 0; i < 4; ++i) {
            const int e = (tid + i * THREADS) * 4;
            // x is streamed exactly once -> nontemporal, keep L2 for the table
            float4v v = __builtin_nontemporal_load((const float4v*)(src + e));
            *(float4v*)(dst + e) = v;
        }
#endif
    }
    __syncthreads();

    // ---------------- Phase 1: wave32 ballot bit-pack of indices ----------------
    // wave w owns batch row w. Lane = tg*8 + c; one ballot packs 4 tables.
    {
        const int wave = tid >> 5;       // 0..7  -> row in tile
        const int lane = tid & 31;
        const int c    = lane & 7;       // bit position within a table
        const int tg   = lane >> 3;      // table subgroup 0..3
        const float* xr = xs[wave];
        #pragma unroll
        for (int k = 0; k < 4; ++k) {
            const int t  = k * 4 + tg;
            const int ia = aa[k * 32 + lane];     // == aa[t*C + c], coalesced
            const int ib = ab[k * 32 + lane];
            const bool pred = xr[ia] > xr[ib];
            const unsigned long long m = __ballot((int)pred);  // wave32: low 32 bits
            if (c == 0)
                sidx[wave][t] = (int)((m >> (8 * tg)) & 0xFFull);
        }
    }
    __syncthreads();

    // ---------------- Phase 2: gather + accumulate (L2-resident table) ----------------
    // thread -> (half r2, float4 column j); 128 float4 groups cover OUT=512.
    {
        const int j  = tid & 127;
        const int r2 = tid >> 7;         // 0..1
        const float4v* tb = (const float4v*)table;
        #pragma unroll
        for (int rr = 0; rr < 4; ++rr) {
            const int row = r2 * 4 + rr;
            const int* si = sidx[row];   // uniform per wave -> LDS broadcast
            float4v acc = {0.f, 0.f, 0.f, 0.f};
            #pragma unroll
            for (int t = 0; t < T_SZ; ++t) {
                const size_t rbase = ((size_t)(t * ROWS_SZ + si[t])) * (OUT_SZ / 4);
                if (rr == 0 && t + 1 < T_SZ) {
                    // hint the next gathered row (global_prefetch_b8)
                    const size_t nbase =
                        ((size_t)((t + 1) * ROWS_SZ + si[t + 1])) * (OUT_SZ / 4);
                    __builtin_prefetch((const void*)(tb + nbase + j), 0, 3);
                }
                acc += tb[rbase + j];    // coalesced 128-bit loads, L2 hit
            }
            float4v* op = (float4v*)y + ((size_t)(b0 + row)) * (OUT_SZ / 4) + j;
            __builtin_nontemporal_store(acc, op);  // y never re-read: bypass caches
        }
    }
}

extern "C" void kernel_launch(void* const* d_in, const int* in_sizes, int n_in,
                              void* d_out, int out_size, void* d_ws, size_t ws_size,
                              hipStream_t stream) {
    (void)in_sizes; (void)n_in; (void)out_size; (void)d_ws; (void)ws_size;
    const float* x     = (const float*)d_in[0];
    const float* table = (const float*)d_in[1];
    const int*   aa    = (const int*)d_in[2];
    const int*   ab    = (const int*)d_in[3];
    float*       y     = (float*)d_out;

    dim3 grid(B_SZ / RB);   // 2048 blocks
    dim3 block(THREADS);    // 8 wave32 waves
    hipLaunchKernelGGL(lut_block_kernel, grid, block, 0, stream, x, table, aa, ab, y);
}